// HashEncoding_35175782154254
// MI455X (gfx1250) — compile-verified
//
#include <hip/hip_runtime.h>
#include <cmath>
#include <stdint.h>

// Instant-NGP hash-grid encoding for MI455X (gfx1250, wave32).
// N=262144 pts, L=16 levels, F=2 feats, 2^19 entries/level (64MB table -> L2-resident).
// Bottleneck: random b64 gathers from L2 (VMEM issue). Strategy:
//   * 8 independent gathers in flight per thread (ILP)
//   * coords staged via CDNA5 async global->LDS DMA (frees VMEM issue slots)
//   * non-temporal output stores (keep the 64MB table resident in 192MB L2)
//   * speculative global_prefetch_b8 on the input stream

#define L_LEVELS 16
#define LOG2_T   19
#define TMASK    ((1u << LOG2_T) - 1u)
#define P1       2654435761u
#define P2       805459861u

typedef float v2f __attribute__((ext_vector_type(2)));
typedef float v4f __attribute__((ext_vector_type(4)));

struct ScaleArgs { float s[L_LEVELS]; };

// Corner selection order matches reference SEL (1 -> ceil, 0 -> floor):
// c: (x,y,z) = (1,1,1)(1,0,1)(0,0,1)(0,1,1)(1,1,0)(1,0,0)(0,0,0)(0,1,0)
__constant__ const int kSelX[8] = {1, 1, 0, 0, 1, 1, 0, 0};
__constant__ const int kSelY[8] = {1, 0, 0, 1, 1, 0, 0, 1};
__constant__ const int kSelZ[8] = {1, 1, 1, 1, 0, 0, 0, 0};

__global__ __launch_bounds__(256)
void hashenc_kernel(const float* __restrict__ in,      // [N,3]
                    const v2f*  __restrict__ table,    // [L*T] float2
                    v2f*        __restrict__ out,      // [N*L] float2
                    ScaleArgs sa, int npts)
{
    // 16 points per 256-thread block. Coords staged by the async DMA engine
    // into LDS, padded to 16B/point so one ds_load_b128 fetches x,y,z.
    __shared__ __align__(16) float s_coords[16 * 4];

    const int tid = threadIdx.x;
    const int t   = blockIdx.x * 256 + tid;
    const int l   = tid & 15;            // lane-minor = level -> coalesced output
    const int n   = t >> 4;
    const int n0  = (blockIdx.x * 256) >> 4;   // first point of this block

    // ---- CDNA5 async global->LDS stage: 48 coord dwords, scattered into the
    // padded LDS layout (VDST carries a per-lane LDS byte address). ----
    {
        const int gidx = n0 * 3 + tid;   // global float index being fetched
        if (tid < 48 && gidx < npts * 3) {
            const uint32_t ldst = (uint32_t)(uintptr_t)(&s_coords[0])
                                + (uint32_t)((tid / 3) * 16 + (tid % 3) * 4);
            const uint32_t voff = (uint32_t)tid * 4u;
            const float*   base = in + (size_t)n0 * 3;
            asm volatile("global_load_async_to_lds_b32 %0, %1, %2"
                         :: "v"(ldst), "v"(voff), "s"(base)
                         : "memory");
        }
#if __has_builtin(__builtin_amdgcn_s_wait_asynccnt)
        __builtin_amdgcn_s_wait_asynccnt(0);
#else
        asm volatile("s_wait_asynccnt 0x0" ::: "memory");
#endif
    }
    __syncthreads();

    if (n >= npts) return;

    // One speculative prefetch per block, ~128 points (1.5KB) ahead of the
    // input stream. SYS-scope speculative prefetch is OOB-safe (silently
    // dropped on translation failure).
    if (tid == 0) __builtin_prefetch(in + (size_t)(n0 + 128) * 3, 0, 0);

    // Coords from LDS: single 16B-aligned ds_load_b128 (w is pad).
    const v4f c4 = ((const v4f*)s_coords)[tid >> 4];
    const float x = c4.x, y = c4.y, z = c4.z;

    const float s  = sa.s[l];
    const float sx = x * s, sy = y * s, sz = z * s;

    const float fx = floorf(sx), fy = floorf(sy), fz = floorf(sz);
    const float cx = ceilf(sx),  cy = ceilf(sy),  cz = ceilf(sz);

    const float ox = sx - fx, oy = sy - fy, oz = sz - fz;
    const float mx = 1.0f - ox, my = 1.0f - oy, mz = 1.0f - oz;

    // uint32 wraparound hash matches reference (int32 -> uint32 cast).
    const unsigned hx_f = (unsigned)(int)fx;         // * PRIMES[0] == 1
    const unsigned hx_c = (unsigned)(int)cx;
    const unsigned hy_f = (unsigned)(int)fy * P1;
    const unsigned hy_c = (unsigned)(int)cy * P1;
    const unsigned hz_f = (unsigned)(int)fz * P2;
    const unsigned hz_c = (unsigned)(int)cz * P2;

    const unsigned base = (unsigned)l << LOG2_T;

    unsigned idx[8];
    float    w[8];
#pragma unroll
    for (int c = 0; c < 8; ++c) {
        const unsigned hx = kSelX[c] ? hx_c : hx_f;
        const unsigned hy = kSelY[c] ? hy_c : hy_f;
        const unsigned hz = kSelZ[c] ? hz_c : hz_f;
        idx[c] = ((hx ^ hy ^ hz) & TMASK) + base;
        const float wx = kSelX[c] ? ox : mx;
        const float wy = kSelY[c] ? oy : my;
        const float wz = kSelZ[c] ? oz : mz;
        w[c] = wx * wy * wz;
    }

    // Issue all 8 gathers before consuming -> 8 b64 loads in flight.
    v2f f[8];
#pragma unroll
    for (int c = 0; c < 8; ++c) f[c] = table[idx[c]];

    v2f acc;
    acc.x = 0.0f; acc.y = 0.0f;
#pragma unroll
    for (int c = 0; c < 8; ++c) {
        acc.x = fmaf(w[c], f[c].x, acc.x);
        acc.y = fmaf(w[c], f[c].y, acc.y);
    }

    // Non-temporal store: output is write-once/streamed; keep the 64MB hash
    // table resident in L2. out element t = n*16 + l -> floats [n*32+2l, +1].
    __builtin_nontemporal_store(acc, &out[t]);
}

extern "C" void kernel_launch(void* const* d_in, const int* in_sizes, int n_in,
                              void* d_out, int out_size, void* d_ws, size_t ws_size,
                              hipStream_t stream) {
    const float* in    = (const float*)d_in[0];   // [N,3] float32
    const float* table = (const float*)d_in[1];   // [L*T, 2] float32

    const int npts = in_sizes[0] / 3;

    // Exact same double-precision formula as the reference numpy code:
    // floor(16 * pow(exp((log(1024)-log(16))/15), i))
    ScaleArgs sa;
    const double g = exp((log(1024.0) - log(16.0)) / 15.0);
    for (int i = 0; i < L_LEVELS; ++i)
        sa.s[i] = (float)floor(16.0 * pow(g, (double)i));

    const int total  = npts * L_LEVELS;
    const int blocks = (total + 255) / 256;

    hashenc_kernel<<<blocks, 256, 0, stream>>>(
        in, (const v2f*)table, (v2f*)d_out, sa, npts);
}